// MessageBuildingLayerLSH_30288109371692
// MI455X (gfx1250) — compile-verified
//
#include <hip/hip_runtime.h>

#define Bsz   16
#define Nn    12800
#define FM    128
#define FN    256
#define BINSZ 128
#define NBINS 100
#define NKEY  200   // keys are in [0, 198]
#define NH    50    // n_bins // 2

typedef __attribute__((ext_vector_type(2))) float v2f;
typedef __attribute__((ext_vector_type(8))) float v8f;

// ---------------------------------------------------------------------------
// Kernel 1: LSH projection + argmax over [mul, -mul]  -> bin_idx per node
// ---------------------------------------------------------------------------
__global__ __launch_bounds__(256) void k_binidx(const float* __restrict__ x_msg,
                                                const int* __restrict__ msk,
                                                const float* __restrict__ rot,
                                                int* __restrict__ binidx) {
    __shared__ float rotS[FM * NH];
    const int tid = threadIdx.x;
    for (int i = tid; i < FM * NH; i += 256) {
        int f = i / NH, h = i % NH;
        rotS[i] = rot[f * (2 * NBINS / 2) + h];  // rot is (128, 100) row-major, use h<50
    }
    __syncthreads();

    const int node = blockIdx.x * 256 + tid;     // grid sized exactly B*N/256
    const float* xr = x_msg + (size_t)node * FM;

    float acc[NH];
#pragma unroll
    for (int h = 0; h < NH; ++h) acc[h] = 0.f;

    for (int f = 0; f < FM; ++f) {
        float xv = xr[f];
        const float* rr = &rotS[f * NH];
#pragma unroll
        for (int h = 0; h < NH; ++h) acc[h] = fmaf(xv, rr[h], acc[h]);
    }

    // argmax over cmul = [acc, -acc], first-max-wins (strict >)
    float best = -INFINITY;
    int arg = 0;
#pragma unroll
    for (int j = 0; j < 2 * NH; ++j) {
        float v = (j < NH) ? acc[j] : -acc[j - NH];
        if (v > best) { best = v; arg = j; }
    }
    int mk = msk[node];
    binidx[node] = arg + (mk != 0 ? 0 : (NBINS - 1));
}

// ---------------------------------------------------------------------------
// Kernel 2: per-batch STABLE counting sort of bin_idx -> bins_split
// 64 threads, each owns a contiguous chunk of 200 indices (stability).
// ---------------------------------------------------------------------------
__global__ __launch_bounds__(64) void k_sort(const int* __restrict__ binidx,
                                             int* __restrict__ sorted,
                                             float* __restrict__ out_bins) {
    __shared__ int hist[64 * NKEY];
    __shared__ int total[NKEY];
    __shared__ int base[NKEY];

    const int b = blockIdx.x, t = threadIdx.x;
    const int chunk = Nn / 64;  // 200
    int* h = &hist[t * NKEY];
    for (int k = 0; k < NKEY; ++k) h[k] = 0;

    const int* bi = binidx + (size_t)b * Nn + t * chunk;
    for (int e = 0; e < chunk; ++e) h[bi[e]]++;
    __syncthreads();

    // cross-thread exclusive prefix per key; total per key
    for (int k = t; k < NKEY; k += 64) {
        int run = 0;
        for (int tt = 0; tt < 64; ++tt) {
            int c = hist[tt * NKEY + k];
            hist[tt * NKEY + k] = run;
            run += c;
        }
        total[k] = run;
    }
    __syncthreads();

    if (t == 0) {
        int s = 0;
        for (int k = 0; k < NKEY; ++k) { base[k] = s; s += total[k]; }
    }
    __syncthreads();

    for (int k = 0; k < NKEY; ++k) h[k] += base[k];
    __syncthreads();

    int*   so = sorted   + (size_t)b * Nn;
    float* fo = out_bins + (size_t)b * Nn;
    for (int e = 0; e < chunk; ++e) {
        int idx = t * chunk + e;
        int k   = bi[e];
        int pos = h[k]++;          // private to (t,k): no races, stable
        so[pos] = idx;
        fo[pos] = (float)idx;
    }
}

// ---------------------------------------------------------------------------
// Kernel 3: per-(batch, bin): gather, x_node copy, Gram via f32 WMMA, kernelize
// ---------------------------------------------------------------------------
__global__ __launch_bounds__(256) void k_bins(const float* __restrict__ x_msg,
                                              const float* __restrict__ x_node,
                                              const int* __restrict__ msk,
                                              const int* __restrict__ sorted,
                                              float* __restrict__ out_node,
                                              float* __restrict__ out_dm,
                                              float* __restrict__ out_msk) {
    constexpr int STR = 132;  // row stride in floats; 132 % 64 = 4 -> conflict-free frags
    __shared__ float A[BINSZ * STR];
    __shared__ float naS[BINSZ];
    __shared__ float mS[BINSZ];
    __shared__ int   idxS[BINSZ];

    const int tid = threadIdx.x;
    const int b   = blockIdx.x / NBINS;
    const int bin = blockIdx.x % NBINS;
    const size_t binbase = (size_t)(b * NBINS + bin);

    if (tid < BINSZ) {
        int idx = sorted[(size_t)b * Nn + bin * BINSZ + tid];
        idxS[tid] = idx;
        float mv = (msk[(size_t)b * Nn + idx] != 0) ? 1.f : 0.f;
        mS[tid] = mv;
        out_msk[binbase * BINSZ + tid] = mv;
    }
    __syncthreads();

    if (tid < BINSZ) {
        // stage masked A row into LDS + row norm
        const int r = tid;
        const float* src = x_msg + ((size_t)b * Nn + idxS[r]) * FM;
        const float mv = mS[r];
        float s = 0.f;
        for (int f = 0; f < FM; f += 4) {
            float4 v = *(const float4*)(src + f);
            v.x *= mv; v.y *= mv; v.z *= mv; v.w *= mv;
            A[r * STR + f + 0] = v.x;
            A[r * STR + f + 1] = v.y;
            A[r * STR + f + 2] = v.z;
            A[r * STR + f + 3] = v.w;
            s += v.x * v.x + v.y * v.y + v.z * v.z + v.w * v.w;
        }
        naS[r] = s;
    } else {
        // concurrently stream the x_node gather (pure bandwidth)
        const int r = tid - BINSZ;
        const float4* src = (const float4*)(x_node + ((size_t)b * Nn + idxS[r]) * FN);
        float4* dst = (float4*)(out_node + (binbase * BINSZ + r) * FN);
        for (int q = 0; q < FN / 4; ++q) dst[q] = src[q];
    }
    __syncthreads();

    // Gram(128x128, K=128) = A * A^T via V_WMMA_F32_16X16X4_F32.
    // Wave w owns the 16-row strip i0 = w*16; sweeps 8 column tiles.
    const int wave = tid >> 5;
    const int lane = tid & 31;
    const int i0   = wave * 16;
    const int rA   = i0 + (lane & 15);
    const int ksel = (lane >> 4) * 2;         // ISA 16x4 f32 A layout: K0/K1 | K2/K3
    const int rowOff = (lane < 16) ? 0 : 8;   // ISA 16x16 f32 C layout

    v2f af[32];
#pragma unroll
    for (int k4 = 0; k4 < 32; ++k4)
        af[k4] = *(const v2f*)&A[rA * STR + k4 * 4 + ksel];

    float* dmb = out_dm + binbase * (BINSZ * BINSZ);
    for (int tj = 0; tj < 8; ++tj) {
        const int j0 = tj * 16;
        const int rB = j0 + (lane & 15);
        v8f c = {0.f, 0.f, 0.f, 0.f, 0.f, 0.f, 0.f, 0.f};
#pragma unroll
        for (int k4 = 0; k4 < 32; ++k4) {
            v2f bf = *(const v2f*)&A[rB * STR + k4 * 4 + ksel];
            c = __builtin_amdgcn_wmma_f32_16x16x4_f32(
                    false, af[k4], false, bf, (short)0, c, false, false);
        }
#pragma unroll
        for (int r = 0; r < 8; ++r) {
            int row = i0 + r + rowOff;
            int col = j0 + (lane & 15);
            float g  = c[r];
            float d2 = naS[row] - 2.f * g + naS[col];
            float D  = sqrtf(fmaxf(d2, 1e-6f));
            float dm = fminf(__expf(-0.1f * D), 1.f);
            dm *= mS[row] * mS[col];
            dmb[row * BINSZ + col] = dm;
        }
    }
}

// ---------------------------------------------------------------------------
extern "C" void kernel_launch(void* const* d_in, const int* in_sizes, int n_in,
                              void* d_out, int out_size, void* d_ws, size_t ws_size,
                              hipStream_t stream) {
    const float* x_msg  = (const float*)d_in[0];
    const float* x_node = (const float*)d_in[1];
    const int*   msk    = (const int*)d_in[2];
    const float* rot    = (const float*)d_in[3];

    float* out = (float*)d_out;
    const size_t BN = (size_t)Bsz * Nn;            // 204800
    float* out_bins = out;                          // (B,100,128)        204800
    float* out_node = out_bins + BN;                // (B,100,128,256)  52428800
    float* out_dm   = out_node + BN * FN;           // (B,100,128,128)  26214400
    float* out_msk  = out_dm   + BN * BINSZ;        // (B,100,128,1)      204800

    int* binidx = (int*)d_ws;        // BN ints
    int* sorted = binidx + BN;       // BN ints

    k_binidx<<<(int)(BN / 256), 256, 0, stream>>>(x_msg, msk, rot, binidx);
    k_sort<<<Bsz, 64, 0, stream>>>(binidx, sorted, out_bins);
    k_bins<<<Bsz * NBINS, 256, 0, stream>>>(x_msg, x_node, msk, sorted,
                                            out_node, out_dm, out_msk);
}